// Model_87058987090235
// MI455X (gfx1250) — compile-verified
//
#include <hip/hip_runtime.h>
#include <math.h>

typedef float v2f __attribute__((ext_vector_type(2)));
typedef float v8f __attribute__((ext_vector_type(8)));

#define NCOLORS 16
#define NPTS 64
#define RTHRESH 0.3f
#define RTHRESH2 0.09f
#define REFINE_BAND 1e-5f
#define WAVES_PER_BLOCK 8
#define VERTS_PER_BLOCK (32 * WAVES_PER_BLOCK)   // 32 verts per wave

__global__ __launch_bounds__(256, 1) void iso_color_kernel(
    const float* __restrict__ pointsSphere,   // (64,2) theta,phi
    const float* __restrict__ colors,         // (16,3)
    const float* __restrict__ isoVert,        // (N,3) unit vectors
    float* __restrict__ out,                  // (N,3)
    int nverts)
{
    // pts[p] = (-2px, -2py, -2pz, |p|^2) so A·[vx,vy,vz,1] = |p|^2 - 2 p·v
    __shared__ float4 sPts[NPTS];

    const int tid = threadIdx.x;
    if (tid < NPTS) {
        float th = pointsSphere[2 * tid + 0];
        float ph = pointsSphere[2 * tid + 1];
        float sth = sinf(th), cth = cosf(th);
        float sph = sinf(ph), cph = cosf(ph);
        float x = sth * cph, y = sth * sph, z = cth;
        sPts[tid] = make_float4(-2.0f * x, -2.0f * y, -2.0f * z, x * x + y * y + z * z);
    }
    __syncthreads();

    const int lane = tid & 31;
    const int n    = lane & 15;
    const bool hi  = lane >= 16;
    const int wave = tid >> 5;
    const int hibase = hi ? 32 : 0;

    const long long vbase = ((long long)blockIdx.x * WAVES_PER_BLOCK + wave) * 32;
    const long long vid   = vbase + lane;
    const long long vld   = vid < (long long)nverts ? vid : (long long)(nverts - 1);

    const float vx = isoVert[vld * 3 + 0];
    const float vy = isoVert[vld * 3 + 1];
    const float vz = isoVert[vld * 3 + 2];

    // Partner coords (lane ^ 16).
    const float ox = __shfl_xor(vx, 16, 32);
    const float oy = __shfl_xor(vy, 16, 32);
    const float oz = __shfl_xor(vz, 16, 32);

    // Per-SET vertex as seen by THIS lane's slice of the C tiles:
    //  acc0 rows on lane L belong to set0 vertex (L&15) -> low lanes: own, high: partner
    //  acc1 rows on lane L belong to set1 vertex (L&15) -> low: partner, high: own
    const float v0x = hi ? ox : vx, v0y = hi ? oy : vy, v0z = hi ? oz : vz;
    const float v1x = hi ? vx : ox, v1y = hi ? vy : oy, v1z = hi ? vz : oz;
    const float v0n2 = v0x * v0x + v0y * v0y + v0z * v0z;
    const float v1n2 = v1x * v1x + v1y * v1y + v1z * v1z;

    // B tiles (4x16 KxN): lanes 0-15 supply K=0,1; lanes 16-31 supply K=2,3.
    v2f b0, b1;
    b0.x = hi ? v0z : v0x;   b0.y = hi ? 1.0f : v0y;
    b1.x = hi ? v1z : v1x;   b1.y = hi ? 1.0f : v1y;

    // A tiles: row M of group g carries point (M<8 ? g*8+M : 32+g*8+M-8).
    // => C layout gives low lanes points 0..31 in order, high lanes 32..63.
    v8f acc0[4], acc1[4];
#pragma unroll
    for (int g = 0; g < 4; ++g) {
        const int pidx = (n < 8) ? (g * 8 + n) : (32 + g * 8 + (n - 8));
        float4 p = sPts[pidx];
        v2f a;
        a.x = hi ? p.z : p.x;
        a.y = hi ? p.w : p.y;
        v8f cz = {};
        acc0[g] = __builtin_amdgcn_wmma_f32_16x16x4_f32(
            false, a, false, b0, (short)0, cz, false, false);
        acc1[g] = __builtin_amdgcn_wmma_f32_16x16x4_f32(
            false, a, false, b1, (short)0, cz, false, false);
    }

    // ---- Fused pass over both sets (each lane: its 32 local points, in order) ----
    // Loop A: materialize s[k] = |p_k - v|^2 (expanded), track threshold proximity.
    float sv0[32], sv1[32];
    float band = 1e30f;
#pragma unroll
    for (int k = 0; k < 32; ++k) {
        float s0 = fmaxf(acc0[k >> 3][k & 7] + v0n2, 0.0f);  // clamp: d~0 => hit
        float s1 = fmaxf(acc1[k >> 3][k & 7] + v1n2, 0.0f);
        sv0[k] = s0;
        sv1[k] = s1;
        band = fminf(band, fminf(fabsf(s0 - RTHRESH2), fabsf(s1 - RTHRESH2)));
    }
    // Single rare branch: expanded form carries ~5e-7 abs error (cancellation at
    // magnitude ~2). If any value sits within 1e-5 of the threshold, recompute
    // component-wise (reference-identical) with the correct per-set vertex.
    if (band < REFINE_BAND) {
#pragma unroll
        for (int k = 0; k < 32; ++k) {
            float4 pp = sPts[hibase + k];        // (-2px,-2py,-2pz,|p|^2)
            float hx = 0.5f * pp.x, hy = 0.5f * pp.y, hz = 0.5f * pp.z; // exact -px,-py,-pz
            float dx0 = hx + v0x, dy0 = hy + v0y, dz0 = hz + v0z;
            float dx1 = hx + v1x, dy1 = hy + v1y, dz1 = hz + v1z;
            sv0[k] = dx0 * dx0 + dy0 * dy0 + dz0 * dz0;
            sv1[k] = dx1 * dx1 + dy1 * dy1 + dz1 * dz1;
        }
    }
    // Loop B: sequential affine blend, both sets interleaved (8 indep chains).
    // Track e = col - b instead of b:   e <- d'*e + (c_{k+1} - c_k)
    // (one FMA/channel; delta-colors are lane-uniform scalars). Miss => d'=1,
    // a ~1-ulp identity. Start: b=0 => e = c_0. End: b = c_0 - e  (32 % 16 == 0).
    float m0 = 1.0f, m1 = 1.0f;
    float e0r = colors[0], e0g = colors[1], e0b = colors[2];
    float e1r = e0r,       e1g = e0g,       e1b = e0b;
#pragma unroll
    for (int k = 0; k < 32; ++k) {
        float d0 = __builtin_amdgcn_sqrtf(sv0[k]);   // raw v_sqrt_f32
        float d1 = __builtin_amdgcn_sqrtf(sv1[k]);
        const int ci = (k & 15) * 3;
        const int cn = ((k + 1) & 15) * 3;
        const float dcr = colors[cn + 0] - colors[ci + 0];  // uniform (SGPR) deltas
        const float dcg = colors[cn + 1] - colors[ci + 1];
        const float dcb = colors[cn + 2] - colors[ci + 2];
        float dp0 = (d0 < RTHRESH) ? d0 : 1.0f;
        float dp1 = (d1 < RTHRESH) ? d1 : 1.0f;
        m0  = dp0 * m0;          m1  = dp1 * m1;
        e0r = dp0 * e0r + dcr;   e1r = dp1 * e1r + dcr;
        e0g = dp0 * e0g + dcg;   e1g = dp1 * e1g + dcg;
        e0b = dp0 * e0b + dcb;   e1b = dp1 * e1b + dcb;
    }
    const float b0r = colors[0] - e0r, b0g = colors[1] - e0g, b0b = colors[2] - e0b;
    const float b1r = colors[0] - e1r, b1g = colors[1] - e1g, b1b = colors[2] - e1b;

    // Stitch halves with a single symmetric 4-float exchange (lane ^ 16):
    //  - low lane n : own b0 = c_mid(set0 vtx n); needs hi's (m0,b0rgb) = 2nd-half affine
    //  - high lane n: own (m1,b1) = 2nd-half affine; needs lo's b1rgb = c_mid(set1 vtx n)
    float p0 = hi ? m0  : b1r;
    float p1 = hi ? b0r : b1g;
    float p2 = hi ? b0g : b1b;
    float p3 = hi ? b0b : 0.0f;
    float q0 = __shfl_xor(p0, 16, 32);
    float q1 = __shfl_xor(p1, 16, 32);
    float q2 = __shfl_xor(p2, 16, 32);
    float q3 = __shfl_xor(p3, 16, 32);

    const float outr = hi ? (m1 * q0 + b1r) : (q0 * b0r + q1);
    const float outg = hi ? (m1 * q1 + b1g) : (q0 * b0g + q2);
    const float outb = hi ? (m1 * q2 + b1b) : (q0 * b0b + q3);

    if (vid < (long long)nverts) {
        out[vid * 3 + 0] = outr;
        out[vid * 3 + 1] = outg;
        out[vid * 3 + 2] = outb;
    }
}

extern "C" void kernel_launch(void* const* d_in, const int* in_sizes, int n_in,
                              void* d_out, int out_size, void* d_ws, size_t ws_size,
                              hipStream_t stream) {
    const float* pointsSphere = (const float*)d_in[0];  // 64*2 f32
    const float* colors       = (const float*)d_in[1];  // 16*3 f32
    const float* isoVert      = (const float*)d_in[2];  // N*3 f32
    float* out = (float*)d_out;

    const int nverts = in_sizes[2] / 3;
    const int blocks = (nverts + VERTS_PER_BLOCK - 1) / VERTS_PER_BLOCK;
    iso_color_kernel<<<blocks, 256, 0, stream>>>(pointsSphere, colors, isoVert, out, nverts);
}